// position_attention_5669356830999
// MI455X (gfx1250) — compile-verified
//
#include <hip/hip_runtime.h>
#include <hip/hip_bf16.h>

#define SB 1024   // sequence length S
#define DB 1024   // hidden dim D
#define AB 256    // attention dim A
#define NSEG 16   // segments for two-level scan (64 positions each)

typedef float v2f __attribute__((ext_vector_type(2)));
typedef float v8f __attribute__((ext_vector_type(8)));

// ---------------------------------------------------------------------------
// Kernel 1: scores[b,s] = sum_a tanh( hidden[b,s,:] @ W1[:,a] + b1[a] ) * v[a]
// 256 blocks (64 rows each of the flattened 16384 rows), 512 thr = 16 waves.
// Wave w owns N-tile [16w, 16w+16) and 4 stacked M-subtiles -> each B-fragment
// feeds 4 WMMAs. K is processed in 8 slices of 128 staged in 32 KB LDS.
// ---------------------------------------------------------------------------
__global__ __launch_bounds__(512) void k_scores(const float* __restrict__ hidden,
                                                const float* __restrict__ W1,
                                                const float* __restrict__ b1,
                                                const float* __restrict__ v,
                                                float* __restrict__ scores) {
  __shared__ float lds_h[64 * 128];   // 32 KB: 64 rows x 128 K-slice
  __shared__ float partial[64];

  const int tid  = threadIdx.x;
  const int tile = blockIdx.x;        // 64 rows per block
  const int wave = tid >> 5;
  const int lane = tid & 31;
  const int half = lane >> 4;         // 0: lanes 0-15, 1: lanes 16-31
  const int idx  = lane & 15;
  const int n0   = wave * 16;

  if (tid < 64) partial[tid] = 0.0f;

  v8f acc[4] = {};
  for (int ks = 0; ks < 8; ++ks) {
    __syncthreads();                  // protects partial[] zeroing / LDS reuse
    // stage 64x128 f32 tile of hidden into LDS (float4, fully coalesced)
    #pragma unroll
    for (int j = 0; j < 4; ++j) {
      const int f   = tid + j * 512;        // float4 index, 2048 total
      const int row = f >> 5;               // 32 float4 per row
      const int kp  = (f & 31) << 2;
      const float4 val =
          *(const float4*)&hidden[((size_t)(tile * 64 + row)) * DB + ks * 128 + kp];
      *(float4*)&lds_h[row * 128 + kp] = val;
    }
    __syncthreads();

    for (int k4 = 0; k4 < 32; ++k4) {
      const int kb_l = k4 * 4 + 2 * half;   // K base within LDS slice (even -> 8B aligned)
      const int kb_g = ks * 128 + kb_l;     // global K base
      // B-frag (4x16 f32): lane {N=idx}, VGPR0/1 = W1[kb][n], W1[kb+1][n]
      v2f b;
      b.x = W1[(size_t)kb_g * AB + n0 + idx];
      b.y = W1[(size_t)(kb_g + 1) * AB + n0 + idx];
      // 4 stacked M-subtiles share this B-fragment
      #pragma unroll
      for (int sub = 0; sub < 4; ++sub) {
        // A-frag (16x4 f32): lane {M=idx}, VGPR0/1 = K=kb, kb+1
        const float2 av = *(const float2*)&lds_h[(sub * 16 + idx) * 128 + kb_l];
        v2f a; a.x = av.x; a.y = av.y;
        acc[sub] = __builtin_amdgcn_wmma_f32_16x16x4_f32(false, a, false, b,
                                                         (short)0, acc[sub],
                                                         false, false);
      }
    }
  }

  // epilogue: tanh(acc + b1)*v, reduce over the 16 N columns of this wave
  const float bb = b1[n0 + idx];
  const float vv = v[n0 + idx];
  #pragma unroll
  for (int sub = 0; sub < 4; ++sub) {
    #pragma unroll
    for (int i = 0; i < 8; ++i) {     // C/D: vgpr i, M = sub*16 + i + 8*half, N = idx
      float t = tanhf(acc[sub][i] + bb) * vv;
      t += __shfl_xor(t, 1, 32);
      t += __shfl_xor(t, 2, 32);
      t += __shfl_xor(t, 4, 32);
      t += __shfl_xor(t, 8, 32);      // bits 0-3: stays within each 16-lane half
      if (idx == 0) atomicAdd(&partial[sub * 16 + i + 8 * half], t);
    }
  }
  __syncthreads();
  if (tid < 64) scores[tile * 64 + tid] = partial[tid];
}

// ---------------------------------------------------------------------------
// Kernel 2: per batch: max over S scores, e_l = exp(s_l - max), segment sums
// of e (16 segments of 64) and batch total. 16 blocks x 256 threads (4 l each).
// ---------------------------------------------------------------------------
__global__ __launch_bounds__(256) void k_softmax_stats(const float* __restrict__ scores,
                                                       float* __restrict__ e,
                                                       float* __restrict__ segden,
                                                       float* __restrict__ total) {
  const int b   = blockIdx.x;
  const int tid = threadIdx.x;
  __shared__ float red[256];
  __shared__ float seg[NSEG];

  float s[4];
  #pragma unroll
  for (int i = 0; i < 4; ++i) s[i] = scores[b * SB + tid * 4 + i];

  float m = fmaxf(fmaxf(s[0], s[1]), fmaxf(s[2], s[3]));
  red[tid] = m;
  __syncthreads();
  for (int off = 128; off > 0; off >>= 1) {
    if (tid < off) red[tid] = fmaxf(red[tid], red[tid + off]);
    __syncthreads();
  }
  m = red[0];
  __syncthreads();

  float le = 0.0f;
  #pragma unroll
  for (int i = 0; i < 4; ++i) {
    const float ev = expf(s[i] - m);
    e[b * SB + tid * 4 + i] = ev;
    le += ev;
  }
  red[tid] = le;                      // thread t covers l in [4t,4t+4) -> seg = t/16
  __syncthreads();
  if (tid < NSEG) {
    float ss = 0.0f;
    for (int k = 0; k < 16; ++k) ss += red[tid * 16 + k];
    seg[tid] = ss;
    segden[b * NSEG + tid] = ss;
  }
  __syncthreads();
  if (tid == 0) {
    float t = 0.0f;
    for (int k = 0; k < NSEG; ++k) t += seg[k];
    total[b] = t;
  }
}

// ---------------------------------------------------------------------------
// Kernel 3a: segment numerators: segnum[b,seg,d] = sum_{l in seg} e_l * h[b,l,d]
// 1024 blocks x 256 threads (b x 16 segs x 4 d-chunks).
// ---------------------------------------------------------------------------
__global__ __launch_bounds__(256) void k_segnum(const float* __restrict__ hidden,
                                                const float* __restrict__ e,
                                                float* __restrict__ segnum) {
  const int id  = blockIdx.x;
  const int b   = id >> 6;
  const int seg = (id >> 2) & 15;
  const int dc  = id & 3;
  const int d   = dc * 256 + threadIdx.x;
  __shared__ float els[64];
  if (threadIdx.x < 64) els[threadIdx.x] = e[b * SB + seg * 64 + threadIdx.x];
  __syncthreads();
  float sn = 0.0f;
  const float* hp = hidden + ((size_t)b * SB + seg * 64) * DB + d;
  for (int i = 0; i < 64; ++i) sn += els[i] * hp[(size_t)i * DB];
  segnum[((size_t)b * NSEG + seg) * DB + d] = sn;
}

// ---------------------------------------------------------------------------
// Kernel 3b: final pooled output via segmented prefix/suffix scan (or total-
// minus-self for cloze). Same grid mapping as 3a.
// ---------------------------------------------------------------------------
__global__ __launch_bounds__(256) void k_pool(const float* __restrict__ hidden,
                                              const float* __restrict__ e,
                                              const float* __restrict__ segnum,
                                              const float* __restrict__ segden,
                                              const float* __restrict__ total,
                                              const int* __restrict__ pt,
                                              float* __restrict__ out) {
  const int id  = blockIdx.x;
  const int b   = id >> 6;
  const int seg = (id >> 2) & 15;
  const int dc  = id & 3;
  const int d   = dc * 256 + threadIdx.x;
  __shared__ float els[64];
  __shared__ float sden[NSEG];
  if (threadIdx.x < 64) els[threadIdx.x] = e[b * SB + seg * 64 + threadIdx.x];
  if (threadIdx.x < NSEG) sden[threadIdx.x] = segden[b * NSEG + threadIdx.x];
  __syncthreads();

  const int type = pt[b];
  const float* hp  = hidden + ((size_t)b * SB + seg * 64) * DB + d;
  float*       op  = out    + ((size_t)b * SB + seg * 64) * DB + d;
  const float* snp = segnum + (size_t)b * NSEG * DB + d;

  if (type == 0) {                    // prefix mask: l <= j, cnt = j+1
    float num = 0.0f, den = 0.0f;
    for (int sg = 0; sg < seg; ++sg) { num += snp[(size_t)sg * DB]; den += sden[sg]; }
    for (int i = 0; i < 64; ++i) {
      const int j = seg * 64 + i;
      num += els[i] * hp[(size_t)i * DB];
      den += els[i];
      op[(size_t)i * DB] = num / (den * (float)(j + 1));
    }
  } else if (type == 1) {             // postfix mask: l >= j, cnt = S-j
    float num = 0.0f, den = 0.0f;
    for (int sg = seg + 1; sg < NSEG; ++sg) { num += snp[(size_t)sg * DB]; den += sden[sg]; }
    for (int i = 63; i >= 0; --i) {
      const int j = seg * 64 + i;
      num += els[i] * hp[(size_t)i * DB];
      den += els[i];
      op[(size_t)i * DB] = num / (den * (float)(SB - j));
    }
  } else {                            // cloze mask: l != j, cnt = S-1
    float totn = 0.0f;
    for (int sg = 0; sg < NSEG; ++sg) totn += snp[(size_t)sg * DB];
    const float totd = total[b];
    for (int i = 0; i < 64; ++i) {
      const float h = hp[(size_t)i * DB];
      op[(size_t)i * DB] = (totn - els[i] * h) / ((totd - els[i]) * (float)(SB - 1));
    }
  }
}

// ---------------------------------------------------------------------------
extern "C" void kernel_launch(void* const* d_in, const int* in_sizes, int n_in,
                              void* d_out, int out_size, void* d_ws, size_t ws_size,
                              hipStream_t stream) {
  const float* hidden = (const float*)d_in[0];   // (16,1024,1024)
  const float* W1     = (const float*)d_in[1];   // (1024,256)
  const float* b1     = (const float*)d_in[2];   // (256,)
  const float* v      = (const float*)d_in[3];   // (256,)
  const int*   pt     = (const int*)  d_in[4];   // (16,)
  float* out = (float*)d_out;                    // (16,1024,1024)

  float* ws        = (float*)d_ws;
  float* w_e       = ws;                 // 16384 : exp(scores - max_b)
  float* w_scores  = ws + 16384;         // 16384 : raw scores
  float* w_segden  = ws + 32768;         // 256   : per-segment sum of e
  float* w_total   = ws + 33024;         // 16    : per-batch sum of e
  float* w_segnum  = ws + 33056;         // 262144: per-segment sum of e*h  (~1.2 MB total)

  k_scores<<<256, 512, 0, stream>>>(hidden, W1, b1, v, w_scores);
  k_softmax_stats<<<16, 256, 0, stream>>>(w_scores, w_e, w_segden, w_total);
  k_segnum<<<1024, 256, 0, stream>>>(hidden, w_e, w_segnum);
  k_pool<<<1024, 256, 0, stream>>>(hidden, w_e, w_segnum, w_segden, w_total, pt, out);
}